// FastAttention_47090021433765
// MI455X (gfx1250) — compile-verified
//
#include <hip/hip_runtime.h>
#include <math.h>

// Shapes fixed by the reference: q,k,v [4,16,4096,64] f32, proj [256,64] f32.
#define BH_ 64
#define N_  4096
#define D_  64
#define M_  256

typedef float v2f __attribute__((ext_vector_type(2)));
typedef float v8f __attribute__((ext_vector_type(8)));

__device__ __constant__ float DATA_NORM = 0.35355339059327373f; // 64^-0.25
__device__ __constant__ float DIAG_C    = 0.0625f;              // 0.5*dn^2
__device__ __constant__ float RATIO     = 0.0625f;              // 256^-0.5
__device__ __constant__ float EPS_      = 1e-4f;

// D = A(16x4 f32) * B(4x16 f32) + C(16x16 f32), full fp32 precision.
__device__ __forceinline__ v8f wmma4(v2f a, v2f b, v8f c) {
  return __builtin_amdgcn_wmma_f32_16x16x4_f32(false, a, false, b, (short)0, c,
                                               false, false);
}

// ---- CDNA5 async global->LDS copy (ASYNCcnt-tracked, bypasses VGPRs) ----
// vdst VGPR carries the wave-relative LDS byte offset (low 32 bits of the
// generic pointer in the LDS aperture); vaddr is the 64-bit global address.
__device__ __forceinline__ void async_g2l_b128(const float* __restrict__ g,
                                               const float* l) {
  unsigned lofs = (unsigned)(uintptr_t)l;
  asm volatile("global_load_async_to_lds_b128 %0, %1, off"
               :: "v"(lofs), "v"(g) : "memory");
}
__device__ __forceinline__ void wait_async0() {
#if __has_builtin(__builtin_amdgcn_s_wait_asynccnt)
  __builtin_amdgcn_s_wait_asynccnt(0);
#else
  asm volatile("s_wait_asynccnt 0x0" ::: "memory");
#endif
}

// ---- order-preserving float<->uint encoding for global atomic max ----
__device__ __forceinline__ unsigned encf(float f) {
  unsigned u = __float_as_uint(f);
  return (u & 0x80000000u) ? ~u : (u | 0x80000000u);
}
__device__ __forceinline__ float decf(unsigned e) {
  unsigned u = (e & 0x80000000u) ? (e & 0x7FFFFFFFu) : ~e;
  return __uint_as_float(u);
}

// Stage DATA_NORM*proj [256,64] into LDS, pair-interleaved so a B-fragment
// (k,k+1 at one column m) is one 8B LDS load: plds[(k>>1)*512 + m*2 + (k&1)].
// Folding the scale here lets k/q/v be async-copied raw.
__device__ __forceinline__ void stage_proj(float* plds,
                                           const float* __restrict__ proj,
                                           int tid) {
  for (int i = tid; i < M_ * D_; i += 256) {
    int m = i >> 6, k = i & 63;
    plds[(k >> 1) * 512 + m * 2 + (k & 1)] = DATA_NORM * proj[i];
  }
}

// One 16x16 tile of xd = X[rows] . (dn*proj)^T : K-dim = 64 -> 16 WMMAs.
// xlds holds raw x as [row][64].
__device__ __forceinline__ v8f xd_tile(const float* xlds, const float* plds,
                                       int rowBase, int mt, int lane) {
  const int half = lane >> 4;
  const int lm = lane & 15;
  const int m = mt * 16 + lm;
  v8f acc = {};
#pragma unroll
  for (int kb = 0; kb < 64; kb += 4) {
    int kk = kb + half * 2;
    v2f a = *(const v2f*)(xlds + (rowBase + lm) * 64 + kk);   // A[row][k..k+1]
    v2f b = *(const v2f*)(plds + (kk >> 1) * 512 + m * 2);    // B[k..k+1][m]
    acc = wmma4(a, b, acc);
  }
  return acc;
}

// ---------------- kernel 0: init workspace ----------------
__global__ void fa_init(unsigned* __restrict__ gmax, float* __restrict__ ctx) {
  int i = blockIdx.x * blockDim.x + threadIdx.x;
  if (i == 0) *gmax = 0u;  // encodes below any finite float
  if (i < BH_ * M_ * D_) ctx[i] = 0.0f;
}

// ---------------- kernel 1: global max of xd_k ----------------
// grid (64, 32), block 256. Each block: 128 rows of k for one (b,h).
__global__ void fa_kmax(const float* __restrict__ k,
                        const float* __restrict__ proj,
                        unsigned* __restrict__ gmax) {
  extern __shared__ float smem[];
  float* plds = smem;          // 16384 f
  float* xlds = plds + 16384;  // 8192 f (raw k tile)
  __shared__ float red[8];

  const int tid = threadIdx.x, lane = tid & 31, w = tid >> 5;
  const int bh = blockIdx.x;
  const int n0 = blockIdx.y * 128;

  const float* kp = k + ((size_t)bh * N_ + n0) * D_;
  for (int i = tid; i < (128 * D_) / 4; i += 256)
    async_g2l_b128(kp + i * 4, xlds + i * 4);
  stage_proj(plds, proj, tid);
  wait_async0();
  __syncthreads();

  float lmax = -INFINITY;
  for (int mt = 0; mt < 16; ++mt) {
    v8f acc = xd_tile(xlds, plds, w * 16, mt, lane);
#pragma unroll
    for (int g = 0; g < 8; ++g) lmax = fmaxf(lmax, acc[g]);
  }
#pragma unroll
  for (int off = 16; off; off >>= 1) lmax = fmaxf(lmax, __shfl_xor(lmax, off));
  if (lane == 0) red[w] = lmax;
  __syncthreads();
  if (tid == 0) {
    float bm = red[0];
    for (int i = 1; i < 8; ++i) bm = fmaxf(bm, red[i]);
    atomicMax(gmax, encf(bm));
  }
}

// ---------------- kernel 2: context = phi(K)^T . V ----------------
// grid (64, 8), block 256. Each block: 4 chunks of 128 rows; atomicAdd once.
__global__ void fa_context(const float* __restrict__ k,
                           const float* __restrict__ v,
                           const float* __restrict__ proj,
                           const unsigned* __restrict__ gmaxp,
                           float* __restrict__ ctx) {
  extern __shared__ float smem[];
  float* plds = smem;           // 16384 f
  float* xlds = plds + 16384;   // 8192 f  (raw k tile)
  float* vlds = xlds + 8192;    // 8192 f  (raw v tile)
  float* flds = vlds + 8192;    // 32768 f (feat tile [128][256])
  float* dlds = flds + 32768;   // 128 f   (diag)

  const int tid = threadIdx.x, lane = tid & 31, w = tid >> 5;
  const int half = lane >> 4, lm = lane & 15;
  const int bh = blockIdx.x;
  const float gmax = decf(*gmaxp);

  stage_proj(plds, proj, tid);

  v8f cacc[8];
#pragma unroll
  for (int t = 0; t < 8; ++t) cacc[t] = (v8f){};

  for (int it = 0; it < 4; ++it) {
    const int n0 = (blockIdx.y * 4 + it) * 128;
    const float* kp = k + ((size_t)bh * N_ + n0) * D_;
    const float* vp = v + ((size_t)bh * N_ + n0) * D_;
    __syncthreads();  // previous chunk's readers done before restage
    for (int i = tid; i < (128 * D_) / 4; i += 256) {
      async_g2l_b128(kp + i * 4, xlds + i * 4);
      async_g2l_b128(vp + i * 4, vlds + i * 4);
    }
    wait_async0();
    __syncthreads();
    if (tid < 128) {  // diag = 0.0625 * sum(x^2)
      float s = 0.0f;
      for (int j = 0; j < D_; ++j) { float x = xlds[tid * 64 + j]; s += x * x; }
      dlds[tid] = DIAG_C * s;
    }
    __syncthreads();

    // GEMM1 + exp transform -> feat tile in LDS (D->A transpose via LDS)
    for (int mt = 0; mt < 16; ++mt) {
      v8f acc = xd_tile(xlds, plds, w * 16, mt, lane);
#pragma unroll
      for (int g = 0; g < 8; ++g) {
        int row = w * 16 + g + half * 8;
        float f = RATIO * (__expf(acc[g] - dlds[row] - gmax) + EPS_);
        flds[row * 256 + mt * 16 + lm] = f;
      }
    }
    __syncthreads();

    // GEMM2: ctx[m][e] += feat^T . V over this 128-row chunk
#pragma unroll
    for (int t = 0; t < 8; ++t) {
      const int mt2 = w * 2 + (t >> 2);
      const int et = t & 3;
      const int m = mt2 * 16 + lm;
      const int e = et * 16 + lm;
      v8f a8 = cacc[t];
      for (int kb = 0; kb < 128; kb += 4) {
        int kk = kb + half * 2;
        v2f a; a.x = flds[kk * 256 + m]; a.y = flds[(kk + 1) * 256 + m];
        v2f b; b.x = vlds[kk * 64 + e];  b.y = vlds[(kk + 1) * 64 + e];
        a8 = wmma4(a, b, a8);
      }
      cacc[t] = a8;
    }
  }

#pragma unroll
  for (int t = 0; t < 8; ++t) {
    const int mt2 = w * 2 + (t >> 2);
    const int et = t & 3;
#pragma unroll
    for (int g = 0; g < 8; ++g) {
      int m = mt2 * 16 + g + half * 8;
      int e = et * 16 + lm;
      atomicAdd(&ctx[(size_t)bh * (M_ * D_) + m * 64 + e], cacc[t][g]);
    }
  }
}

// ---------------- kernel 3: out = phi(Q) . context ----------------
// grid (64, 64), block 256. Each block: 64 rows of q for one (b,h).
__global__ void fa_out(const float* __restrict__ q,
                       const float* __restrict__ proj,
                       const float* __restrict__ ctx,
                       float* __restrict__ out) {
  extern __shared__ float smem[];
  float* plds = smem;           // 16384 f
  float* clds = plds + 16384;   // 16384 f (context [256][64])
  float* xlds = clds + 16384;   // 4096 f  (raw q tile [64][64])
  float* flds = xlds + 4096;    // 16384 f (xd/feat tile [64][256])
  float* dlds = flds + 16384;   // 64 f
  float* rlds = dlds + 64;      // 64 f (row max)

  const int tid = threadIdx.x, lane = tid & 31, w = tid >> 5;
  const int half = lane >> 4, lm = lane & 15;
  const int bh = blockIdx.x;
  const int n0 = blockIdx.y * 64;

  const float* cp = ctx + (size_t)bh * (M_ * D_);
  const float* qp = q + ((size_t)bh * N_ + n0) * D_;
  for (int i = tid; i < (M_ * D_) / 4; i += 256)
    async_g2l_b128(cp + i * 4, clds + i * 4);
  for (int i = tid; i < (64 * D_) / 4; i += 256)
    async_g2l_b128(qp + i * 4, xlds + i * 4);
  stage_proj(plds, proj, tid);
  wait_async0();
  __syncthreads();

  if (tid < 64) {
    float s = 0.0f;
    for (int j = 0; j < D_; ++j) { float x = xlds[tid * 64 + j]; s += x * x; }
    dlds[tid] = DIAG_C * s;
  }
  __syncthreads();

  // GEMM1: 4 strips x 16 m-tiles = 64 tiles; wave w: strip w>>1, 8 m-tiles.
  {
    const int strip = w >> 1;
    for (int i = 0; i < 8; ++i) {
      const int mt = (w & 1) * 8 + i;
      v8f acc = xd_tile(xlds, plds, strip * 16, mt, lane);
#pragma unroll
      for (int g = 0; g < 8; ++g) {
        int row = strip * 16 + g + half * 8;
        flds[row * 256 + mt * 16 + lm] = acc[g];  // raw xd
      }
    }
  }
  __syncthreads();
  if (tid < 64) {  // per-row stabilizer
    float rm = -INFINITY;
    for (int j = 0; j < M_; ++j) rm = fmaxf(rm, flds[tid * 256 + j]);
    rlds[tid] = rm;
  }
  __syncthreads();
  for (int i = tid; i < 64 * M_; i += 256) {
    int r = i >> 8;
    flds[i] = RATIO * (__expf(flds[i] - dlds[r] - rlds[r]) + EPS_);
  }
  __syncthreads();

  // GEMM2: out[n][e] = feat . context ; 16 tiles, 2 per wave, K = 256.
  float* outp = out + ((size_t)bh * N_ + n0) * D_;
#pragma unroll
  for (int t = 0; t < 2; ++t) {
    const int tile = w * 2 + t;
    const int nt = tile >> 2, et = tile & 3;
    const int n = nt * 16 + lm;
    const int e = et * 16 + lm;
    v8f acc = {};
    for (int kb = 0; kb < M_; kb += 4) {
      int kk = kb + half * 2;
      v2f a = *(const v2f*)(flds + n * 256 + kk);             // feat[n][k..k+1]
      v2f b; b.x = clds[kk * 64 + e]; b.y = clds[(kk + 1) * 64 + e];
      acc = wmma4(a, b, acc);
    }
#pragma unroll
    for (int g = 0; g < 8; ++g) {
      int row = nt * 16 + g + half * 8;
      outp[row * 64 + e] = acc[g];
    }
  }
}

static constexpr int SMEM1 = (16384 + 8192) * 4;                          // 98304
static constexpr int SMEM2 = (16384 + 8192 + 8192 + 32768 + 128) * 4;     // 262656
static constexpr int SMEM3 = (16384 + 16384 + 4096 + 16384 + 64 + 64) * 4;// 213504

extern "C" void kernel_launch(void* const* d_in, const int* in_sizes, int n_in,
                              void* d_out, int out_size, void* d_ws, size_t ws_size,
                              hipStream_t stream) {
  const float* q    = (const float*)d_in[0];
  const float* k    = (const float*)d_in[1];
  const float* v    = (const float*)d_in[2];
  const float* proj = (const float*)d_in[3];
  float* out = (float*)d_out;

  unsigned* gmax = (unsigned*)d_ws;
  float* ctx = (float*)((char*)d_ws + 256);  // 64*256*64 f32 = 4 MB

  (void)hipFuncSetAttribute((const void*)fa_kmax,
                            hipFuncAttributeMaxDynamicSharedMemorySize, SMEM1);
  (void)hipFuncSetAttribute((const void*)fa_context,
                            hipFuncAttributeMaxDynamicSharedMemorySize, SMEM2);
  (void)hipFuncSetAttribute((const void*)fa_out,
                            hipFuncAttributeMaxDynamicSharedMemorySize, SMEM3);

  fa_init<<<dim3((BH_ * M_ * D_ + 255) / 256), dim3(256), 0, stream>>>(gmax, ctx);
  fa_kmax<<<dim3(BH_, N_ / 128), dim3(256), SMEM1, stream>>>(k, proj, gmax);
  fa_context<<<dim3(BH_, N_ / 512), dim3(256), SMEM2, stream>>>(k, v, proj, gmax, ctx);
  fa_out<<<dim3(BH_, N_ / 64), dim3(256), SMEM3, stream>>>(q, proj, ctx, out);
}